// NumberEmbedder_52819507806298
// MI455X (gfx1250) — compile-verified
//
#include <hip/hip_runtime.h>

#define HIDDEN 128
#define ROWS   40                       // DIGITS * 10
#define TABLE_FLOATS (ROWS * HIDDEN)    // 5120 floats = 20 KB

typedef __attribute__((ext_vector_type(4))) float floatx4;
typedef __attribute__((ext_vector_type(4))) int   intx4;

// Pointer types matching the CDNA5 async-copy builtin prototype:
//   void __builtin_amdgcn_global_load_async_to_lds_b128(int4 AS(1)*, int4 AS(3)*, imm, imm)
typedef __attribute__((address_space(1))) intx4* g4ptr_t;
typedef __attribute__((address_space(3))) intx4* l4ptr_t;

__global__ __launch_bounds__(256) void number_embedder_kernel(
    const int*   __restrict__ nums,   // [N], values in [0, 10000)
    const float* __restrict__ emb,    // [40][128] fp32
    float*       __restrict__ out,    // [N][128] fp32
    int N)
{
    __shared__ float tab[TABLE_FLOATS];   // 20 KB of the 320 KB/WGP LDS

    const int tid = threadIdx.x;

    // ---- Stage the embedding table into LDS ----------------------------
    // 256 threads x 16 B = 4 KB per step; 5 steps = 20 KB.
#if __has_builtin(__builtin_amdgcn_global_load_async_to_lds_b128)
    #pragma unroll
    for (int s = 0; s < 5; ++s) {
        const int foff = (s * 256 + tid) * 4;             // float index (x4 per lane)
        g4ptr_t gp = (g4ptr_t)(emb + foff);
        l4ptr_t lp = (l4ptr_t)(tab + foff);
        // CDNA5: GLOBAL_LOAD_ASYNC_TO_LDS_B128 (tracked by ASYNCcnt)
        __builtin_amdgcn_global_load_async_to_lds_b128(gp, lp, 0, 0);
    }
  #if __has_builtin(__builtin_amdgcn_s_wait_asynccnt)
    __builtin_amdgcn_s_wait_asynccnt(0);
  #else
    asm volatile("s_wait_asynccnt 0" ::: "memory");
  #endif
#else
    // Fallback: synchronous copy through VGPRs.
    #pragma unroll
    for (int s = 0; s < 5; ++s) {
        const int foff = (s * 256 + tid) * 4;
        *(floatx4*)&tab[foff] = *(const floatx4*)&emb[foff];
    }
#endif
    __syncthreads();

    // ---- Gather + sum, one wave32 per number ---------------------------
    const int lane   = tid & 31;
    const int wave   = blockIdx.x * (blockDim.x >> 5) + (tid >> 5);
    const int nWaves = gridDim.x * (blockDim.x >> 5);
    const int col    = lane * 4;            // lane's float4 column in [0,128)

    for (int n = wave; n < N; n += nWaves) {
        const int v  = nums[n];             // broadcast load (same addr per wave)
        const int q0 = v / 10;
        const int q1 = v / 100;
        const int q2 = v / 1000;            // v < 10000 -> q2 in [0,9] == digit 3
        const int d0 = v  - q0 * 10;
        const int d1 = q0 - q1 * 10;
        const int d2 = q1 - q2 * 10;

        // Conflict-free ds_load_b128 gathers: row stride 512 B, lanes cover
        // consecutive dwords -> each of the 64 banks hit exactly twice.
        const floatx4 a = *(const floatx4*)&tab[( 0 + d0) * HIDDEN + col];
        const floatx4 b = *(const floatx4*)&tab[(10 + d1) * HIDDEN + col];
        const floatx4 c = *(const floatx4*)&tab[(20 + d2) * HIDDEN + col];
        const floatx4 d = *(const floatx4*)&tab[(30 + q2) * HIDDEN + col];
        const floatx4 s = (a + b) + (c + d);

        // 512 MB write-once stream > 192 MB L2: bypass with non-temporal store.
        __builtin_nontemporal_store(s, (floatx4*)&out[(size_t)n * HIDDEN + col]);
    }
}

extern "C" void kernel_launch(void* const* d_in, const int* in_sizes, int n_in,
                              void* d_out, int out_size, void* d_ws, size_t ws_size,
                              hipStream_t stream) {
    const int*   nums = (const int*)d_in[0];
    const float* emb  = (const float*)d_in[1];
    float*       out  = (float*)d_out;
    const int    N    = in_sizes[0];

    // 4096 blocks x 8 waves = 32768 wave32s, ~30 numbers each at N = 1M.
    const int threads = 256;
    const int blocks  = 4096;
    number_embedder_kernel<<<blocks, threads, 0, stream>>>(nums, emb, out, N);
}